// ConditionalEstimation_30734785970302
// MI455X (gfx1250) — compile-verified
//
#include <hip/hip_runtime.h>

typedef __attribute__((ext_vector_type(2))) float v2f;
typedef __attribute__((ext_vector_type(4))) float v4f;
typedef __attribute__((ext_vector_type(8))) float v8f;

#define DIM     384
#define K2      (DIM * DIM)          // 147456 : K of the context GEMV
#define NTILES  (DIM / 16)           // 24 row-tiles of a
#define SLICES  288                  // K-slices per row-tile
#define CHUNK   (K2 / SLICES)        // 512 K per wave (multiple of 8)
#define NCAND   65536
#define XTILES  (NCAND / 16)         // 4096 row-tiles of x

__device__ __forceinline__ v8f wmma4(v2f a, v2f b, v8f c) {
  // D = A(16x4,f32) * B(4x16,f32) + C(16x16,f32)
  return __builtin_amdgcn_wmma_f32_16x16x4_f32(
      /*neg_a=*/false, a, /*neg_b=*/false, b,
      /*c_mod=*/(short)0, c, /*reuse_a=*/false, /*reuse_b=*/false);
}

// Non-temporal streaming load (data read exactly once; keep L2 for reused vectors)
__device__ __forceinline__ v4f nt_load4(const float* p) {
  return __builtin_nontemporal_load(reinterpret_cast<const v4f*>(p));
}
__device__ __forceinline__ v4f ld4(const float* p) {
  return *reinterpret_cast<const v4f*>(p);
}

// w[j*DIM + k] = y[j] * z[k]
__global__ void init_w_kernel(const float* __restrict__ y,
                              const float* __restrict__ z,
                              float* __restrict__ w) {
  int idx = blockIdx.x * blockDim.x + threadIdx.x;
  if (idx < K2) w[idx] = y[idx / DIM] * z[idx % DIM];
}

// part[(tile*SLICES + s)*16 + m] = sum over this wave's K-chunk of a[row_m] * w
__global__ void __launch_bounds__(256)
contract_kernel(const float* __restrict__ a,
                const float* __restrict__ w,
                float* __restrict__ part) {
  const int lane = threadIdx.x & 31;
  const int wid  = threadIdx.x >> 5;
  const int gw   = blockIdx.x * 8 + wid;      // global wave id
  const int tile = gw / SLICES;
  const int s    = gw % SLICES;
  const int m    = lane & 15;                 // row within tile
  const int h    = lane >> 4;                 // half-wave: K sub-offset
  const float* __restrict__ arow = a + (size_t)(tile * 16 + m) * K2;
  const int k0 = s * CHUNK;

  v8f c = {0.f, 0.f, 0.f, 0.f, 0.f, 0.f, 0.f, 0.f};
  #pragma unroll 4
  for (int k = k0; k < k0 + CHUNK; k += 8) {
    const v4f av = nt_load4(arow + k + 4 * h);   // streamed once: TH=NT
    const v4f bv = ld4(w + k + 4 * h);           // reused by all 24 tiles: RT
    v2f a1 = {av.x, av.y}; v2f b1 = {bv.x, bv.y};
    c = wmma4(a1, b1, c);                        // covers K = k, k+1, k+4, k+5
    v2f a2 = {av.z, av.w}; v2f b2 = {bv.z, bv.w};
    c = wmma4(a2, b2, c);                        // covers K = k+2, k+3, k+6, k+7
  }

  // D layout: VGPR r -> row r (lanes 0-15) / row r+8 (lanes 16-31); all cols equal
  if (m == 0) {
    float* p = part + (size_t)(tile * SLICES + s) * 16 + 8 * h;
    #pragma unroll
    for (int r = 0; r < 8; ++r) __builtin_nontemporal_store(c[r], p + r);
  }
}

// context[i] = deterministic sum over SLICES partials
__global__ void reduce_kernel(const float* __restrict__ part,
                              float* __restrict__ context) {
  int i = blockIdx.x * blockDim.x + threadIdx.x;
  if (i >= DIM) return;
  int tile = i >> 4, m = i & 15;
  const float* p = part + (size_t)tile * SLICES * 16 + m;
  float sum = 0.f;
  for (int s = 0; s < SLICES; ++s) sum += p[(size_t)s * 16];
  context[i] = sum;
}

// scores[n] = (x[n,:] . context) / (context[0] + 1e-8)
__global__ void __launch_bounds__(256)
scores_kernel(const float* __restrict__ x,
              const float* __restrict__ context,
              float* __restrict__ out) {
  const int lane = threadIdx.x & 31;
  const int wid  = threadIdx.x >> 5;
  const int tile = blockIdx.x * 8 + wid;      // 0..XTILES-1
  const int m    = lane & 15;
  const int h    = lane >> 4;
  const float* __restrict__ xrow = x + (size_t)(tile * 16 + m) * DIM;

  v8f c = {0.f, 0.f, 0.f, 0.f, 0.f, 0.f, 0.f, 0.f};
  #pragma unroll 4
  for (int k = 0; k < DIM; k += 8) {
    const v4f av = nt_load4(xrow + k + 4 * h);     // streamed once: TH=NT
    const v4f bv = ld4(context + k + 4 * h);       // reused by all waves: RT
    v2f a1 = {av.x, av.y}; v2f b1 = {bv.x, bv.y};
    c = wmma4(a1, b1, c);
    v2f a2 = {av.z, av.w}; v2f b2 = {bv.z, bv.w};
    c = wmma4(a2, b2, c);
  }

  const float denom = context[0] + 1e-8f;
  if (m == 0) {
    float* o = out + (size_t)tile * 16 + 8 * h;
    #pragma unroll
    for (int r = 0; r < 8; ++r) o[r] = c[r] / denom;
  }
}

extern "C" void kernel_launch(void* const* d_in, const int* in_sizes, int n_in,
                              void* d_out, int out_size, void* d_ws, size_t ws_size,
                              hipStream_t stream) {
  const float* x = (const float*)d_in[0];   // [65536, 384]
  const float* y = (const float*)d_in[1];   // [384]
  const float* z = (const float*)d_in[2];   // [384]
  const float* a = (const float*)d_in[3];   // [384, 384, 384]
  float* out = (float*)d_out;               // [65536]

  // workspace layout (floats): w[K2] | part[NTILES*SLICES*16] | context[DIM]
  float* w       = (float*)d_ws;
  float* part    = w + K2;
  float* context = part + (size_t)NTILES * SLICES * 16;

  init_w_kernel<<<K2 / 256, 256, 0, stream>>>(y, z, w);
  contract_kernel<<<(NTILES * SLICES) / 8, 256, 0, stream>>>(a, w, part);
  reduce_kernel<<<(DIM + 127) / 128, 128, 0, stream>>>(part, context);
  scores_kernel<<<XTILES / 8, 256, 0, stream>>>(x, context, out);
}